// EntanglementFormationModel_11467562680836
// MI455X (gfx1250) — compile-verified
//
#include <hip/hip_runtime.h>
#include <hip/hip_bf16.h>
#include <math.h>

// ---------------------------------------------------------------------------
// Problem constants (from reference)
// ---------------------------------------------------------------------------
#define T_TERMS 131072
#define RNK     64          // rank = dimA*dimB = 64
#define DIMA    8
#define DIMB    8

// ws layout (float offsets)
#define WS_SRE      0
#define WS_SIM      4096
#define WS_WRE      8192
#define WS_WIM      12288
#define WS_GRE      16384
#define WS_GIM      20480
#define WS_NS       24576                  // 8 * 4096 Newton-Schulz temps
#define WS_PART     57344                  // 64 blocks * 8192 gram partials
#define GRAM_BLOCKS 64
#define WS_CONTRIB  (WS_PART + GRAM_BLOCKS*8192)   // 8192 per-tile contributions
#define NUM_TILES   (T_TERMS/16)           // 8192

typedef float v2f __attribute__((ext_vector_type(2)));
typedef float v8f __attribute__((ext_vector_type(8)));

// fp32 WMMA: D(16x16) = A(16x4) * B(4x16) + C   (V_WMMA_F32_16X16X4_F32)
__device__ __forceinline__ v8f wmma4(v2f a, v2f b, v8f c) {
  return __builtin_amdgcn_wmma_f32_16x16x4_f32(false, a, false, b,
                                               (short)0, c, false, false);
}

// ---------------------------------------------------------------------------
// Kernel 1: Gram matrix partials.  S = X^H X,  X = theta_re + i theta_im.
// Sre = Ar^T Ar + Ai^T Ai ;  Sim = Ar^T Ai - Ai^T Ar.
// 64 blocks x 512 threads (16 waves). Wave w owns output tile (w>>2, w&3).
// Each block loops over 2048/64 = 32 chunks of 64 rows staged in LDS.
// Deterministic: per-block partial tiles stored to ws, reduced later.
// ---------------------------------------------------------------------------
__global__ __launch_bounds__(512) void gram_kernel(
    const float* __restrict__ thr, const float* __restrict__ thi,
    float* __restrict__ partials) {
  __shared__ float xr[64 * 64];
  __shared__ float xi[64 * 64];

  const int tid  = threadIdx.x;
  const int lane = tid & 31;
  const int wave = tid >> 5;               // 0..15
  const int i0   = (wave >> 2) * 16;       // output row tile
  const int j0   = (wave & 3) * 16;        // output col tile
  const int m    = lane & 15;              // M index for A-frag / N for B-frag
  const int kk   = (lane >> 4) << 1;       // K base: 0 or 2

  v8f accRe = {};
  v8f accI1 = {};   // Ar^T Ai
  v8f accI2 = {};   // Ai^T Ar

  const int chunksPerBlock = (T_TERMS / 64) / gridDim.x;
  for (int cc = 0; cc < chunksPerBlock; ++cc) {
    const int t0 = (blockIdx.x * chunksPerBlock + cc) * 64;
    __syncthreads();
    for (int e = tid; e < 64 * 64; e += blockDim.x) {
      xr[e] = thr[t0 * 64 + e];
      xi[e] = thi[t0 * 64 + e];
    }
    __syncthreads();

    for (int k = 0; k < 64; k += 4) {
      v2f aR, aI, bR, bI;
      aR.x = xr[(k + kk) * 64 + i0 + m];  aR.y = xr[(k + kk + 1) * 64 + i0 + m];
      aI.x = xi[(k + kk) * 64 + i0 + m];  aI.y = xi[(k + kk + 1) * 64 + i0 + m];
      bR.x = xr[(k + kk) * 64 + j0 + m];  bR.y = xr[(k + kk + 1) * 64 + j0 + m];
      bI.x = xi[(k + kk) * 64 + j0 + m];  bI.y = xi[(k + kk + 1) * 64 + j0 + m];
      accRe = wmma4(aR, bR, accRe);
      accRe = wmma4(aI, bI, accRe);
      accI1 = wmma4(aR, bI, accI1);
      accI2 = wmma4(aI, bR, accI2);
    }
  }

  // C/D layout: VGPR v -> M = v + 8*(lane>=16), N = lane&15
  float* pre = partials + (size_t)blockIdx.x * 8192;
  float* pim = pre + 4096;
  const int rofs = 8 * (lane >> 4);
  const int col  = lane & 15;
#pragma unroll
  for (int v = 0; v < 8; ++v) {
    const int r = i0 + v + rofs;
    const int c = j0 + col;
    pre[r * 64 + c] = accRe[v];
    pim[r * 64 + c] = accI1[v] - accI2[v];
  }
}

__global__ __launch_bounds__(256) void gram_reduce_kernel(
    const float* __restrict__ partials, float* __restrict__ ws) {
  const int e = blockIdx.x * blockDim.x + threadIdx.x;  // 0..8191
  if (e < 4096) {
    float s = 0.f;
    for (int b = 0; b < GRAM_BLOCKS; ++b) s += partials[(size_t)b * 8192 + e];
    ws[WS_SRE + e] = s;
  } else if (e < 8192) {
    const int f = e - 4096;
    float s = 0.f;
    for (int b = 0; b < GRAM_BLOCKS; ++b) s += partials[(size_t)b * 8192 + 4096 + f];
    ws[WS_SIM + f] = s;
  }
}

// ---------------------------------------------------------------------------
// Kernel 2: one block computes W = S^{-1/2} via scaled Newton-Schulz and
// Gm[r][p] = sqrt(EVL[r]) * conj(EVC[p][r]).  All matrices live in ws
// (global); __syncthreads gives block-level global visibility.
// ---------------------------------------------------------------------------
__global__ __launch_bounds__(256) void prep_kernel(
    const float* __restrict__ evl, const float* __restrict__ evc_re,
    const float* __restrict__ evc_im, float* __restrict__ ws) {
  float* Sre = ws + WS_SRE;  float* Sim = ws + WS_SIM;
  float* Wre = ws + WS_WRE;  float* Wim = ws + WS_WIM;
  float* Gre = ws + WS_GRE;  float* Gim = ws + WS_GIM;
  float* Yre = ws + WS_NS;           float* Yim = Yre + 4096;
  float* Zre = Yim + 4096;           float* Zim = Zre + 4096;
  float* Tre = Zim + 4096;           float* Tim = Tre + 4096;
  float* Ure = Tim + 4096;           float* Uim = Ure + 4096;

  const int tid = threadIdx.x;
  __shared__ float s_scale;
  if (tid == 0) {
    float trace = 0.f;
    for (int i = 0; i < 64; ++i) trace += Sre[i * 64 + i];
    s_scale = 64.0f / trace;         // 1/c, c = mean eigenvalue scale
  }
  __syncthreads();
  const float scale = s_scale;

  for (int e = tid; e < 4096; e += 256) {
    const int r = e >> 6, c = e & 63;
    Yre[e] = Sre[e] * scale;  Yim[e] = Sim[e] * scale;
    Zre[e] = (r == c) ? 1.f : 0.f;  Zim[e] = 0.f;
  }
  __threadfence(); __syncthreads();

  for (int it = 0; it < 14; ++it) {
    // T = 1.5 I - 0.5 * Z*Y   (complex)
    for (int e = tid; e < 4096; e += 256) {
      const int r = e >> 6, c = e & 63;
      float sr = 0.f, si = 0.f;
      for (int k = 0; k < 64; ++k) {
        const float zr = Zre[r * 64 + k], zi = Zim[r * 64 + k];
        const float yr = Yre[k * 64 + c], yi = Yim[k * 64 + c];
        sr += zr * yr - zi * yi;
        si += zr * yi + zi * yr;
      }
      Tre[e] = ((r == c) ? 1.5f : 0.f) - 0.5f * sr;
      Tim[e] = -0.5f * si;
    }
    __threadfence(); __syncthreads();
    // Y = Y*T
    for (int e = tid; e < 4096; e += 256) {
      const int r = e >> 6, c = e & 63;
      float sr = 0.f, si = 0.f;
      for (int k = 0; k < 64; ++k) {
        const float yr = Yre[r * 64 + k], yi = Yim[r * 64 + k];
        const float tr = Tre[k * 64 + c], ti = Tim[k * 64 + c];
        sr += yr * tr - yi * ti;
        si += yr * ti + yi * tr;
      }
      Ure[e] = sr; Uim[e] = si;
    }
    __threadfence(); __syncthreads();
    for (int e = tid; e < 4096; e += 256) { Yre[e] = Ure[e]; Yim[e] = Uim[e]; }
    __threadfence(); __syncthreads();
    // Z = T*Z
    for (int e = tid; e < 4096; e += 256) {
      const int r = e >> 6, c = e & 63;
      float sr = 0.f, si = 0.f;
      for (int k = 0; k < 64; ++k) {
        const float tr = Tre[r * 64 + k], ti = Tim[r * 64 + k];
        const float zr = Zre[k * 64 + c], zi = Zim[k * 64 + c];
        sr += tr * zr - ti * zi;
        si += tr * zi + ti * zr;
      }
      Ure[e] = sr; Uim[e] = si;
    }
    __threadfence(); __syncthreads();
    for (int e = tid; e < 4096; e += 256) { Zre[e] = Ure[e]; Zim[e] = Uim[e]; }
    __threadfence(); __syncthreads();
  }

  // S^{-1/2} = sqrt(scale) * Z ;  Gm = diag(sqrt(EVL)) * EVC^H
  const float rs = sqrtf(scale);
  for (int e = tid; e < 4096; e += 256) {
    Wre[e] = Zre[e] * rs;
    Wim[e] = Zim[e] * rs;
  }
  for (int e = tid; e < 4096; e += 256) {
    const int r = e >> 6, p = e & 63;
    const float sq = sqrtf(evl[r]);
    Gre[e] =  sq * evc_re[p * 64 + r];
    Gim[e] = -sq * evc_im[p * 64 + r];
  }
}

// ---------------------------------------------------------------------------
// Kernel 3: fused per-tile pipeline. Block = 128 threads = 4 waves; each wave
// owns a 16-term tile:
//   theta1 = X*W (WMMA) -> LDS -> prob -> psi = theta1*Gm (WMMA) -> LDS ->
//   per-lane 8x8 complex Jacobi eigensolve -> entropy -> per-tile contrib.
// ---------------------------------------------------------------------------
__global__ __launch_bounds__(128) void fused_kernel(
    const float* __restrict__ thr, const float* __restrict__ thi,
    const float* __restrict__ evl, const float* __restrict__ ws,
    float* __restrict__ contribs) {
  __shared__ float t_re[4][16 * 64];
  __shared__ float t_im[4][16 * 64];
  __shared__ float s_prob[4][16];

  const float* Wre = ws + WS_WRE;  const float* Wim = ws + WS_WIM;
  const float* Gre = ws + WS_GRE;  const float* Gim = ws + WS_GIM;

  const int tid  = threadIdx.x;
  const int lane = tid & 31;
  const int wave = tid >> 5;                 // 0..3
  const int tile = blockIdx.x * 4 + wave;    // 0..8191
  const int t0   = tile * 16;
  const int m    = lane & 15;
  const int kk   = (lane >> 4) << 1;
  const int rofs = 8 * (lane >> 4);
  const int col  = lane & 15;

  // ---- GEMM1: theta1 = X * W (complex, W Hermitian) ----
  v8f accRe[4] = {};
  v8f accIm[4] = {};
  for (int k = 0; k < 64; k += 4) {
    v2f aR, aI;
    aR.x = thr[(size_t)(t0 + m) * 64 + k + kk];
    aR.y = thr[(size_t)(t0 + m) * 64 + k + kk + 1];
    aI.x = thi[(size_t)(t0 + m) * 64 + k + kk];
    aI.y = thi[(size_t)(t0 + m) * 64 + k + kk + 1];
#pragma unroll
    for (int j = 0; j < 4; ++j) {
      const int j0 = j * 16;
      v2f bR, bI, bIn;
      bR.x = Wre[(k + kk) * 64 + j0 + m];  bR.y = Wre[(k + kk + 1) * 64 + j0 + m];
      bI.x = Wim[(k + kk) * 64 + j0 + m];  bI.y = Wim[(k + kk + 1) * 64 + j0 + m];
      bIn = -bI;
      accRe[j] = wmma4(aR, bR, accRe[j]);
      accRe[j] = wmma4(aI, bIn, accRe[j]);   // re -= Xim*Wim
      accIm[j] = wmma4(aR, bI, accIm[j]);
      accIm[j] = wmma4(aI, bR, accIm[j]);
    }
  }
#pragma unroll
  for (int j = 0; j < 4; ++j)
#pragma unroll
    for (int v = 0; v < 8; ++v) {
      const int M = v + rofs;
      const int N = j * 16 + col;
      t_re[wave][M * 64 + N] = accRe[j][v];
      t_im[wave][M * 64 + N] = accIm[j][v];
    }
  __syncthreads();

  // ---- prob_m = sum_r |theta1|^2 EVL[r] ----
  if (lane < 16) {
    float p = 0.f;
    for (int r = 0; r < 64; ++r) {
      const float x = t_re[wave][lane * 64 + r];
      const float y = t_im[wave][lane * 64 + r];
      p += (x * x + y * y) * evl[r];
    }
    s_prob[wave][lane] = p;
  }
  __syncthreads();

  // ---- GEMM2: psi = theta1 * Gm ----
  v8f pRe[4] = {};
  v8f pIm[4] = {};
  for (int k = 0; k < 64; k += 4) {
    v2f aR, aI;
    aR.x = t_re[wave][m * 64 + k + kk];  aR.y = t_re[wave][m * 64 + k + kk + 1];
    aI.x = t_im[wave][m * 64 + k + kk];  aI.y = t_im[wave][m * 64 + k + kk + 1];
#pragma unroll
    for (int j = 0; j < 4; ++j) {
      const int j0 = j * 16;
      v2f bR, bI, bIn;
      bR.x = Gre[(k + kk) * 64 + j0 + m];  bR.y = Gre[(k + kk + 1) * 64 + j0 + m];
      bI.x = Gim[(k + kk) * 64 + j0 + m];  bI.y = Gim[(k + kk + 1) * 64 + j0 + m];
      bIn = -bI;
      pRe[j] = wmma4(aR, bR, pRe[j]);
      pRe[j] = wmma4(aI, bIn, pRe[j]);
      pIm[j] = wmma4(aR, bI, pIm[j]);
      pIm[j] = wmma4(aI, bR, pIm[j]);
    }
  }
  __syncthreads();
#pragma unroll
  for (int j = 0; j < 4; ++j)
#pragma unroll
    for (int v = 0; v < 8; ++v) {
      const int M = v + rofs;
      const int N = j * 16 + col;
      t_re[wave][M * 64 + N] = pRe[j][v];
      t_im[wave][M * 64 + N] = pIm[j][v];
    }
  __syncthreads();

  // ---- per-term: rdm = psi psi^H / prob, Jacobi eigenvalues, entropy ----
  float contrib = 0.f;
  if (lane < 16) {
    const float prob = s_prob[wave][lane];
    const float invp = 1.0f / prob;
    const float* pr = &t_re[wave][lane * 64];
    const float* pi = &t_im[wave][lane * 64];

    float Ar[8][8], Ai[8][8];
    for (int a = 0; a < 8; ++a)
      for (int c2 = 0; c2 < 8; ++c2) {
        float rr = 0.f, ri = 0.f;
        for (int b = 0; b < 8; ++b) {
          const float x1 = pr[a * 8 + b],  y1 = pi[a * 8 + b];
          const float x2 = pr[c2 * 8 + b], y2 = pi[c2 * 8 + b];
          rr += x1 * x2 + y1 * y2;        // Re(psi_a conj(psi_c))
          ri += y1 * x2 - x1 * y2;        // Im(psi_a conj(psi_c))
        }
        Ar[a][c2] = rr * invp;
        Ai[a][c2] = ri * invp;
      }

    // cyclic complex Jacobi (phase-absorbed Givens), eigenvalues only
    for (int sweep = 0; sweep < 7; ++sweep) {
      for (int p = 0; p < 7; ++p) {
        for (int q = p + 1; q < 8; ++q) {
          const float apr = Ar[p][q], api = Ai[p][q];
          const float mag2 = apr * apr + api * api;
          if (mag2 < 1e-26f) continue;
          const float mag = sqrtf(mag2);
          const float ur = apr / mag, ui = api / mag;   // e^{i phi}
          const float app = Ar[p][p], aqq = Ar[q][q];
          const float tau = (aqq - app) / (2.f * mag);
          const float tt  = (tau >= 0.f ? 1.f : -1.f) /
                            (fabsf(tau) + sqrtf(1.f + tau * tau));
          const float cth = 1.0f / sqrtf(1.f + tt * tt);
          const float sg  = tt * cth;                   // s = sg * e^{i phi}
          // columns: a'_kp = c a_kp - conj(s) a_kq ; a'_kq = s a_kp + c a_kq
          for (int k = 0; k < 8; ++k) {
            const float xr = Ar[k][p], xi2 = Ai[k][p];
            const float yr = Ar[k][q], yi2 = Ai[k][q];
            const float cyr = sg * (ur * yr + ui * yi2);
            const float cyi = sg * (ur * yi2 - ui * yr);
            const float sxr = sg * (ur * xr - ui * xi2);
            const float sxi = sg * (ur * xi2 + ui * xr);
            Ar[k][p] = cth * xr - cyr;  Ai[k][p] = cth * xi2 - cyi;
            Ar[k][q] = sxr + cth * yr;  Ai[k][q] = sxi + cth * yi2;
          }
          // rows: a'_pk = c a_pk - s a_qk ; a'_qk = conj(s) a_pk + c a_qk
          for (int k = 0; k < 8; ++k) {
            const float xr = Ar[p][k], xi2 = Ai[p][k];
            const float yr = Ar[q][k], yi2 = Ai[q][k];
            const float syr = sg * (ur * yr - ui * yi2);
            const float syi = sg * (ur * yi2 + ui * yr);
            const float cxr = sg * (ur * xr + ui * xi2);
            const float cxi = sg * (ur * xi2 - ui * xr);
            Ar[p][k] = cth * xr - syr;  Ai[p][k] = cth * xi2 - syi;
            Ar[q][k] = cxr + cth * yr;  Ai[q][k] = cxi + cth * yi2;
          }
          Ar[p][q] = 0.f; Ai[p][q] = 0.f;
          Ar[q][p] = 0.f; Ai[q][p] = 0.f;
          Ai[p][p] = 0.f; Ai[q][q] = 0.f;
        }
      }
    }

    float ent = 0.f;
    for (int k = 0; k < 8; ++k) {
      const float lam = Ar[k][k];
      ent += lam * logf(fmaxf(lam, 1e-10f));
    }
    contrib = prob * ent;   // final kernel applies the minus sign
  }

  // wave reduction over lanes 0..15
  contrib += __shfl_down(contrib, 8);
  contrib += __shfl_down(contrib, 4);
  contrib += __shfl_down(contrib, 2);
  contrib += __shfl_down(contrib, 1);
  if (lane == 0) contribs[tile] = contrib;
}

// ---------------------------------------------------------------------------
// Final deterministic reduction of per-tile contributions.
// ---------------------------------------------------------------------------
__global__ __launch_bounds__(256) void final_reduce_kernel(
    const float* __restrict__ contribs, float* __restrict__ out) {
  __shared__ float sbuf[256];
  float s = 0.f;
  for (int i = threadIdx.x; i < NUM_TILES; i += 256) s += contribs[i];
  sbuf[threadIdx.x] = s;
  __syncthreads();
  for (int off = 128; off > 0; off >>= 1) {
    if (threadIdx.x < off) sbuf[threadIdx.x] += sbuf[threadIdx.x + off];
    __syncthreads();
  }
  if (threadIdx.x == 0) out[0] = -sbuf[0];
}

// ---------------------------------------------------------------------------
extern "C" void kernel_launch(void* const* d_in, const int* in_sizes, int n_in,
                              void* d_out, int out_size, void* d_ws, size_t ws_size,
                              hipStream_t stream) {
  (void)in_sizes; (void)n_in; (void)out_size; (void)ws_size;
  const float* thr    = (const float*)d_in[0];   // theta_re [T,64]
  const float* thi    = (const float*)d_in[1];   // theta_im [T,64]
  const float* evl    = (const float*)d_in[2];   // EVL [64]
  const float* evc_re = (const float*)d_in[3];   // EVC_re [64,64]
  const float* evc_im = (const float*)d_in[4];   // EVC_im [64,64]
  float* out = (float*)d_out;
  float* ws  = (float*)d_ws;

  gram_kernel<<<GRAM_BLOCKS, 512, 0, stream>>>(thr, thi, ws + WS_PART);
  gram_reduce_kernel<<<32, 256, 0, stream>>>(ws + WS_PART, ws);
  prep_kernel<<<1, 256, 0, stream>>>(evl, evc_re, evc_im, ws);
  fused_kernel<<<NUM_TILES / 4, 128, 0, stream>>>(thr, thi, evl, ws,
                                                  ws + WS_CONTRIB);
  final_reduce_kernel<<<1, 256, 0, stream>>>(ws + WS_CONTRIB, out);
}